// AveragedNormals_90838558310767
// MI455X (gfx1250) — compile-verified
//
#include <hip/hip_runtime.h>
#include <hip/hip_bf16.h>

// ---------------------------------------------------------------------------
// AveragedNormals for MI455X (gfx1250, wave32).
// Distance matrix via V_WMMA_F32_16X16X4_F32 Gram tiles:
//   d2(q,c) = |q|^2 + |c|^2 - 2 * <q,c>,  <q,c> from a 16x16x4 fp32 WMMA.
// Candidate chunks are double-buffered in LDS via GLOBAL_LOAD_ASYNC_TO_LDS_B32
// (ASYNCcnt-tracked), overlapping the DMA of chunk i+1 with WMMA + top-k of
// chunk i.  Falls back to synchronous staging if the builtin is unavailable.
// ---------------------------------------------------------------------------

#define NPTS    8192
#define NBATCH  2
#define KLRF    33
#define KAVG    32
#define QT      128       // queries per block
#define CT      128       // candidates per chunk
#define NCHUNK  (NPTS / CT)
#define DSTRIDE 133       // LDS tile row stride (gcd(133,64)=1 -> conflict free)
#define EPSF    1e-12f

typedef float v2f __attribute__((ext_vector_type(2)));
typedef float v8f __attribute__((ext_vector_type(8)));

#ifndef __has_builtin
#define __has_builtin(x) 0
#endif

#if __has_builtin(__builtin_amdgcn_global_load_async_to_lds_b32)
#define HAVE_ASYNC_LDS 1
#else
#define HAVE_ASYNC_LDS 0
#endif

#if HAVE_ASYNC_LDS
typedef __attribute__((address_space(1))) int gas_int;   // global AS
typedef __attribute__((address_space(3))) int las_int;   // LDS AS

__device__ __forceinline__ void async_copy_b32(const float* g, float* l) {
    __builtin_amdgcn_global_load_async_to_lds_b32(
        (gas_int*)(float*)g,     // drop const, then generic->global AS cast
        (las_int*)l,             // generic->LDS addrspacecast
        /*offset=*/0, /*cpol=*/0);
}
#if __has_builtin(__builtin_amdgcn_s_wait_asynccnt)
#define WAIT_ASYNC(n) __builtin_amdgcn_s_wait_asynccnt(n)
#else
#define WAIT_ASYNC(n) asm volatile("s_wait_asynccnt %0" ::"i"(n) : "memory")
#endif
#endif

// Smallest-eigenvalue unit eigenvector of a symmetric 3x3 (double precision,
// trigonometric eigenvalue formula + row cross-products).
__device__ __forceinline__ void smallest_evec3(
    double a00, double a01, double a02, double a11, double a12, double a22,
    double& vx, double& vy, double& vz)
{
    double q   = (a00 + a11 + a22) * (1.0 / 3.0);
    double b00 = a00 - q, b11 = a11 - q, b22 = a22 - q;
    double p2  = b00 * b00 + b11 * b11 + b22 * b22 +
                 2.0 * (a01 * a01 + a02 * a02 + a12 * a12);
    double lam = q;
    if (p2 > 1e-300) {
        double p   = sqrt(p2 * (1.0 / 6.0));
        double inv = 1.0 / p;
        double d00 = b00 * inv, d11 = b11 * inv, d22 = b22 * inv;
        double e01 = a01 * inv, e02 = a02 * inv, e12 = a12 * inv;
        double detB = d00 * (d11 * d22 - e12 * e12)
                    - e01 * (e01 * d22 - e12 * e02)
                    + e02 * (e01 * e12 - d11 * e02);
        double r = 0.5 * detB;
        r = fmin(1.0, fmax(-1.0, r));
        double phi = acos(r) * (1.0 / 3.0);
        lam = q + 2.0 * p * cos(phi + 2.0943951023931953);  // + 2*pi/3 -> min
    }
    double m00 = a00 - lam, m11 = a11 - lam, m22 = a22 - lam;
    // rows of (A - lam I): r0=(m00,a01,a02) r1=(a01,m11,a12) r2=(a02,a12,m22)
    double x0 = a01 * a12 - a02 * m11;   // r0 x r1
    double y0 = a02 * a01 - m00 * a12;
    double z0 = m00 * m11 - a01 * a01;
    double x1 = a01 * m22 - a02 * a12;   // r0 x r2
    double y1 = a02 * a02 - m00 * m22;
    double z1 = m00 * a12 - a01 * a02;
    double x2 = m11 * m22 - a12 * a12;   // r1 x r2
    double y2 = a12 * a02 - a01 * m22;
    double z2 = a01 * a12 - m11 * a02;
    double n0 = x0 * x0 + y0 * y0 + z0 * z0;
    double n1 = x1 * x1 + y1 * y1 + z1 * z1;
    double n2 = x2 * x2 + y2 * y2 + z2 * z2;
    double bx = x0, by = y0, bz = z0, bn = n0;
    if (n1 > bn) { bx = x1; by = y1; bz = z1; bn = n1; }
    if (n2 > bn) { bx = x2; by = y2; bz = z2; bn = n2; }
    if (bn < 1e-300) { vx = 0.0; vy = 0.0; vz = 1.0; return; }
    double s = 1.0 / sqrt(bn);
    vx = bx * s; vy = by * s; vz = bz * s;
}

__global__ __launch_bounds__(256, 1)
void AveragedNormals_topk_normals(const float* __restrict__ v,
                                  int*   __restrict__ idx_out,
                                  float* __restrict__ nrm_out)
{
    __shared__ float s_dt[QT][DSTRIDE];          // d2 tile, later neigh scratch
    __shared__ float s_topd[QT][KLRF];
    __shared__ int   s_topi[QT][KLRF];
    __shared__ float s_cb[2][CT * 3];            // candidate chunk AoS, 2 bufs
    __shared__ float s_sqq[QT];

    const int b     = blockIdx.y;
    const int qbase = blockIdx.x * QT;
    const int tid   = threadIdx.x;
    const int wv    = tid >> 5;                  // wave id (8 waves)
    const int lane  = tid & 31;
    const int m     = lane & 15;                 // row/col within 16
    const int lh    = lane >> 4;                 // lane half
    const float* vb = v + (size_t)b * NPTS * 3;

    for (int i = tid; i < QT * KLRF; i += 256) {
        (&s_topd[0][0])[i] = 3.0e38f;
        (&s_topi[0][0])[i] = 0;
    }
    if (tid < QT) {
        int q = qbase + tid;
        float x = vb[3 * q], y = vb[3 * q + 1], z = vb[3 * q + 2];
        s_sqq[tid] = x * x + y * y + z * z;
    }

    // A operand: wave wv owns query rows qbase + wv*16 .. +15
    // (ISA 7.12.2, 32-bit A 16x4: lanes 0-15 = (K0,K1), lanes 16-31 = (K2,K3))
    const int qa = qbase + wv * 16 + m;
    const float qax = vb[3 * qa], qay = vb[3 * qa + 1], qaz = vb[3 * qa + 2];
    v2f aop;
    aop.x = lh ? qaz : qax;
    aop.y = lh ? 0.0f : qay;

#if HAVE_ASYNC_LDS
    // prologue: async-stage chunk 0 into buffer 0
    if (tid < CT) {
        const float* src = &vb[3 * tid];
        float*       dst = &s_cb[0][3 * tid];
        async_copy_b32(src + 0, dst + 0);
        async_copy_b32(src + 1, dst + 1);
        async_copy_b32(src + 2, dst + 2);
    }
#endif
    __syncthreads();

    for (int ci = 0; ci < NCHUNK; ++ci) {
        const int cb  = ci * CT;
        const int cur = ci & 1;

#if HAVE_ASYNC_LDS
        if (ci + 1 < NCHUNK) {
            if (tid < CT) {
                const float* src = &vb[3 * (cb + CT + tid)];
                float*       dst = &s_cb[cur ^ 1][3 * tid];
                async_copy_b32(src + 0, dst + 0);
                async_copy_b32(src + 1, dst + 1);
                async_copy_b32(src + 2, dst + 2);
            }
            WAIT_ASYNC(3);   // drain chunk ci (in-order); ci+1 stays in flight
        } else {
            WAIT_ASYNC(0);
        }
#else
        if (tid < CT) {
            const float* src = &vb[3 * (cb + tid)];
            s_cb[cur][3 * tid]     = src[0];
            s_cb[cur][3 * tid + 1] = src[1];
            s_cb[cur][3 * tid + 2] = src[2];
            if (cb + CT < NPTS)
                __builtin_prefetch(&vb[3 * (cb + CT + tid)], 0, 1);
        }
#endif
        __syncthreads();

        // 8 waves x 8 column tiles of 16x16 Gram via WMMA fp32 16x16x4.
        // All LDS loads unconditional -> v_cndmask selects, EXEC untouched.
        const float* cbuf = s_cb[cur];
        #pragma unroll
        for (int t = 0; t < 8; ++t) {
            const int c = t * 16 + m;
            const float px = cbuf[3 * c];
            const float py = cbuf[3 * c + 1];
            const float pz = cbuf[3 * c + 2];
            const float sc = px * px + py * py + pz * pz;  // |c|^2 (col = m)
            v2f bop;
            bop.x = lh ? pz : px;
            bop.y = lh ? 0.0f : py;
            v8f acc = {};
            acc = __builtin_amdgcn_wmma_f32_16x16x4_f32(
                false, aop, false, bop, (short)0, acc, false, false);
            // C/D layout: VGPR r -> row wv*16 + r + 8*lh, col t*16 + m
            #pragma unroll
            for (int r = 0; r < 8; ++r) {
                const int row = wv * 16 + r + 8 * lh;
                float d2 = s_sqq[row] + sc - 2.0f * acc[r];
                s_dt[row][t * 16 + m] = d2;
            }
        }
        __syncthreads();

        // top-33 (ascending d2, stable tie-break by index): 1 thread / row
        if (tid < QT) {
            float* td = s_topd[tid];
            int*   ti = s_topi[tid];
            float worst = td[KLRF - 1];
            for (int j = 0; j < CT; ++j) {
                float d = s_dt[tid][j];
                if (d < worst) {
                    int p = KLRF - 1;
                    while (p > 0 && td[p - 1] > d) {
                        td[p] = td[p - 1]; ti[p] = ti[p - 1]; --p;
                    }
                    td[p] = d;
                    ti[p] = b * NPTS + cb + j;   // global point index
                    worst = td[KLRF - 1];
                }
            }
        }
        __syncthreads();
    }

    // ---- per-point SHOT normal (weighted covariance -> min-eig eigenvector)
    if (tid < QT) {
        const int qi = qbase + tid;
        const int gq = b * NPTS + qi;
        const float qx = vb[3 * qi], qy = vb[3 * qi + 1], qz = vb[3 * qi + 2];
        float* nb = s_dt[tid];                   // reuse: 133 >= 99 floats
        int*   ti = s_topi[tid];

        float radius = 0.0f;
        for (int k = 0; k < KLRF; ++k) {
            int g = ti[k];
            float dx = v[3 * g]     - qx;
            float dy = v[3 * g + 1] - qy;
            float dz = v[3 * g + 2] - qz;
            nb[3 * k] = dx; nb[3 * k + 1] = dy; nb[3 * k + 2] = dz;
            float nn = sqrtf(dx * dx + dy * dy + dz * dz + EPSF);
            radius = fmaxf(radius, nn);
        }
        double c00 = 0, c01 = 0, c02 = 0, c11 = 0, c12 = 0, c22 = 0;
        for (int k = 0; k < KLRF; ++k) {
            float dx = nb[3 * k], dy = nb[3 * k + 1], dz = nb[3 * k + 2];
            float nn = sqrtf(dx * dx + dy * dy + dz * dz + EPSF);
            double w = (double)(radius - nn);
            double X = dx, Y = dy, Z = dz;
            c00 += w * X * X; c01 += w * X * Y; c02 += w * X * Z;
            c11 += w * Y * Y; c12 += w * Y * Z; c22 += w * Z * Z;
        }
        double ex, ey, ez;
        smallest_evec3(c00, c01, c02, c11, c12, c22, ex, ey, ez);

        int nge = 0;
        for (int k = 0; k < KLRF; ++k) {
            double proj = nb[3 * k] * ex + nb[3 * k + 1] * ey + nb[3 * k + 2] * ez;
            nge += (proj >= 0.0) ? 1 : 0;
        }
        if (2 * nge < KLRF) { ex = -ex; ey = -ey; ez = -ez; }

        nrm_out[3 * gq]     = (float)ex;
        nrm_out[3 * gq + 1] = (float)ey;
        nrm_out[3 * gq + 2] = (float)ez;
        for (int k = 0; k < KLRF; ++k)
            idx_out[(size_t)gq * KLRF + k] = ti[k];
    }
}

__global__ __launch_bounds__(256)
void AveragedNormals_avg(const int*   __restrict__ idx,
                         const float* __restrict__ nrm,
                         float*       __restrict__ out)
{
    int i = blockIdx.x * 256 + threadIdx.x;
    if (i >= NBATCH * NPTS) return;
    const int* row = idx + (size_t)i * KLRF;
    float sx = 0.f, sy = 0.f, sz = 0.f;
    for (int k = 0; k < KAVG; ++k) {
        int g = row[k];
        sx += nrm[3 * g]; sy += nrm[3 * g + 1]; sz += nrm[3 * g + 2];
    }
    sx *= (1.0f / KAVG); sy *= (1.0f / KAVG); sz *= (1.0f / KAVG);
    float inv = 1.0f / sqrtf(sx * sx + sy * sy + sz * sz + EPSF);
    out[3 * i]     = sx * inv;
    out[3 * i + 1] = sy * inv;
    out[3 * i + 2] = sz * inv;
}

extern "C" void kernel_launch(void* const* d_in, const int* in_sizes, int n_in,
                              void* d_out, int out_size, void* d_ws, size_t ws_size,
                              hipStream_t stream) {
    (void)in_sizes; (void)n_in; (void)out_size; (void)ws_size;
    const float* v = (const float*)d_in[0];
    float* out = (float*)d_out;

    // workspace layout: [idx: 16384*33 ints][normals: 16384*3 floats]
    int*   idxbuf = (int*)d_ws;
    float* nrmbuf = (float*)((char*)d_ws +
                             (size_t)NBATCH * NPTS * KLRF * sizeof(int));

    dim3 grid1(NPTS / QT, NBATCH);
    AveragedNormals_topk_normals<<<grid1, 256, 0, stream>>>(v, idxbuf, nrmbuf);

    int total = NBATCH * NPTS;
    AveragedNormals_avg<<<(total + 255) / 256, 256, 0, stream>>>(idxbuf, nrmbuf, out);
}